// CascadingSinkCacheCompile_71451075936263
// MI455X (gfx1250) — compile-verified
//
#include <hip/hip_runtime.h>
#include <stdint.h>

// ---------------------------------------------------------------------------
// CascadingSinkCache on MI455X (gfx1250).
//
// Phase 1 (sim_kernel, 1 block): data-independent scalar simulation of the
//   cascade routing. Hot slot->token table lives in LDS; mask/pos written
//   directly to d_out; table exported to d_ws for phase 2.
// Phase 2 (gather_kernel, 65600 blocks): HBM-bound gather / zero-fill of
//   keys/values/sk/sv using gfx1250 async global<->LDS B128 DMA ops
//   (ASYNCcnt path), plus the scores gather.
// ---------------------------------------------------------------------------

namespace {
constexpr int H_       = 32;
constexpr int D_       = 128;
constexpr int WINDOW_  = 512;
constexpr int MAXSEQ_  = 8192;
constexpr int NSINK_   = 4;
constexpr int CASC_    = MAXSEQ_ / WINDOW_;   // 16
constexpr int S_IN_    = 1024;

constexpr long long KV_ELEMS   = (long long)H_ * MAXSEQ_ * D_;   // 33,554,432
constexpr long long SINK_ELEMS = (long long)H_ * NSINK_ * D_;    // 16,384

// Output tuple layout (flat, element units; f32 and i32 are both 4B):
// (keys, values, scores, mask, pos, sk, sv, spos, smask, stored, starts)
constexpr long long OFF_KEYS   = 0;
constexpr long long OFF_VALUES = KV_ELEMS;
constexpr long long OFF_SCORES = 2 * KV_ELEMS;                   // 67,108,864
constexpr long long OFF_MASK   = OFF_SCORES + MAXSEQ_;
constexpr long long OFF_POS    = OFF_MASK   + MAXSEQ_;
constexpr long long OFF_SK     = OFF_POS    + MAXSEQ_;
constexpr long long OFF_SV     = OFF_SK     + SINK_ELEMS;
constexpr long long OFF_SPOS   = OFF_SV     + SINK_ELEMS;
constexpr long long OFF_SMASK  = OFF_SPOS   + NSINK_;
constexpr long long OFF_STORED = OFF_SMASK  + NSINK_;
constexpr long long OFF_STARTS = OFF_STORED + CASC_;

constexpr float NEG_ = -3.4028234663852886e38f;   // jnp.finfo(f32).min

constexpr long long ROWS_K     = (long long)H_ * MAXSEQ_;        // 262,144
constexpr long long ROWS_KV2   = 2 * ROWS_K;                     // 524,288
constexpr long long ROWS_TOTAL = ROWS_KV2 + 2LL * H_ * NSINK_;   // 524,544
constexpr int ROW_BLOCKS    = (int)(ROWS_TOTAL / 8);             // 65,568 (8 rows/block)
constexpr int SCALAR_BLOCKS = (MAXSEQ_ + 255) / 256;             // 32
}

// ---------------------------------------------------------------------------
// Phase 1: exact transcription of the reference's per-step routing logic.
// Invariants used (provable by induction on the reference):
//   * a slot has mask==0.0 iff it was ever written (append sets it; evict /
//     overwrite only hit previously-appended slots because starts moves only
//     once stored==WINDOW),
//   * scores[s] == input_scores[token currently stored at s] (evict carries
//     the stored score along with the stored item),
// so the token table + pos/mask fully determine the outputs.
// ---------------------------------------------------------------------------
__global__ __launch_bounds__(256) void sim_kernel(float* __restrict__ out,
                                                  int* __restrict__ ws_tok) {
  __shared__ int ltok[MAXSEQ_];
  const int tid = threadIdx.x;
  float* out_mask = out + OFF_MASK;
  int*   out_pos  = (int*)out + OFF_POS;

  for (int s = tid; s < MAXSEQ_; s += 256) {
    ltok[s]     = -1;
    out_mask[s] = NEG_;
    out_pos[s]  = 0;
  }
  __syncthreads();

  if (tid == 0) {
    int stored[CASC_], starts[CASC_];
#pragma unroll
    for (int i = 0; i < CASC_; ++i) { stored[i] = 0; starts[i] = 0; }

    int*   out_spos  = (int*)out + OFF_SPOS;
    float* out_smask = out + OFF_SMASK;
    int ssinks = 0;

    for (int t = 0; t < S_IN_; ++t) {
      if (ssinks < NSINK_) {           // add_sink (sk/sv payload copied in phase 2)
        out_spos[ssinks]  = ssinks;
        out_smask[ssinks] = 0.0f;
        ++ssinks;
      } else {                          // add_key
        const int e = t - NSINK_;       // == (seen - 1 - NSINK)
        int ctok = t;
        bool done = false;
        for (int i = 0; i < CASC_ && !done; ++i) {
          const int  l   = i * WINDOW_;
          const bool dci = (e & ((1 << i) - 1)) == 0;   // (e % 2^i) == 0
          if (dci ? (stored[i] < WINDOW_) : (stored[i] == 0)) {
            // append
            const int s = starts[i] + l + stored[i];
            ltok[s] = ctok; out_mask[s] = 0.0f; out_pos[s] = t;
            ++stored[i]; done = true;
          } else if (dci) {
            // evict: write current item, carry old item to next cascade
            const int s  = starts[i] + l;
            const int ot = ltok[s];
            ltok[s] = ctok; out_pos[s] = t;
            starts[i] = (starts[i] + 1) % WINDOW_;
            ctok = ot;                                  // done stays false
          } else {
            // overwrite
            const int s = starts[i] + l;
            ltok[s] = ctok; out_pos[s] = t;
            done = true;
          }
        }
      }
    }
    int* out_stored = (int*)out + OFF_STORED;
    int* out_starts = (int*)out + OFF_STARTS;
#pragma unroll
    for (int i = 0; i < CASC_; ++i) { out_stored[i] = stored[i]; out_starts[i] = starts[i]; }
  }
  __syncthreads();
  for (int s = tid; s < MAXSEQ_; s += 256) ws_tok[s] = ltok[s];
}

// ---------------------------------------------------------------------------
// Phase 2: one 512B row per wave32 (16B per lane). Occupied rows move through
// the CDNA5 async global->LDS->global B128 path; empty rows are zero-filled
// with direct B128 stores. Tail blocks gather scores.
// ---------------------------------------------------------------------------
__global__ __launch_bounds__(256) void gather_kernel(
    const float* __restrict__ keys_in, const float* __restrict__ values_in,
    const float* __restrict__ scores_in, const int* __restrict__ ws_tok,
    float* __restrict__ out) {
  __shared__ float lbuf[8][D_];   // 512B staging per wave, 8 waves/block
  const int bid = blockIdx.x;
  const int tid = threadIdx.x;

  if (bid < ROW_BLOCKS) {
    const int w    = tid >> 5;
    const int lane = tid & 31;
    const long long row = (long long)bid * 8 + w;

    const float* src;
    float*       dst;
    int          tok;
    if (row < ROWS_K) {                                   // keys
      const int h = (int)(row >> 13);
      const int s = (int)(row & (MAXSEQ_ - 1));
      tok = ws_tok[s];
      dst = out + OFF_KEYS + row * D_;
      src = keys_in + ((long long)h * S_IN_ + tok) * D_;
    } else if (row < ROWS_KV2) {                          // values
      const long long r2 = row - ROWS_K;
      const int h = (int)(r2 >> 13);
      const int s = (int)(r2 & (MAXSEQ_ - 1));
      tok = ws_tok[s];
      dst = out + OFF_VALUES + r2 * D_;
      src = values_in + ((long long)h * S_IN_ + tok) * D_;
    } else {                                              // sk / sv rows
      const int r3     = (int)(row - ROWS_KV2);           // 0..255
      const int within = r3 & 127;                        // h*NSINK + j
      const int h = within >> 2;
      const int j = within & 3;
      tok = j;                                            // always occupied
      dst = out + ((r3 < 128) ? OFF_SK : OFF_SV) + (long long)within * D_;
      src = keys_in + ((long long)h * S_IN_ + j) * D_;    // sv <- tk too (ref bug)
    }

    float* d4 = dst + lane * 4;
    if (tok >= 0) {
      const float* s4 = src + lane * 4;
      const unsigned lds_off = (unsigned)(uintptr_t)&lbuf[w][lane * 4];
      // copies 16B per lane from global address s4 to LDS offset lds_off (ASYNCcnt-tracked)
      asm volatile("global_load_async_to_lds_b128 %0, %1, off"
                   :: "v"(lds_off), "v"((unsigned long long)(uintptr_t)s4)
                   : "memory");
      asm volatile("s_wait_asynccnt 0x0" ::: "memory");
      // copies 16B from LDS offset lds_off to global address d4; completion drained by S_ENDPGM's implicit wait-idle
      asm volatile("global_store_async_from_lds_b128 %0, %1, off"
                   :: "v"((unsigned long long)(uintptr_t)d4), "v"(lds_off)
                   : "memory");
    } else {
      float4 z; z.x = 0.f; z.y = 0.f; z.z = 0.f; z.w = 0.f;
      *reinterpret_cast<float4*>(d4) = z;
    }
  } else {
    // scores[s] = input_scores[tok[s]] (stored-score invariant) or 0 if empty
    const int s = (bid - ROW_BLOCKS) * 256 + tid;
    if (s < MAXSEQ_) {
      const int tok = ws_tok[s];
      out[OFF_SCORES + s] = (tok >= 0) ? scores_in[tok] : 0.0f;
    }
  }
}

extern "C" void kernel_launch(void* const* d_in, const int* in_sizes, int n_in,
                              void* d_out, int out_size, void* d_ws, size_t ws_size,
                              hipStream_t stream) {
  (void)in_sizes; (void)n_in; (void)out_size; (void)ws_size;
  const float* keys_in   = (const float*)d_in[0];
  const float* values_in = (const float*)d_in[1];
  const float* scores_in = (const float*)d_in[2];
  float* out    = (float*)d_out;
  int*   ws_tok = (int*)d_ws;      // 8192 ints = 32 KB of scratch

  sim_kernel<<<1, 256, 0, stream>>>(out, ws_tok);
  gather_kernel<<<ROW_BLOCKS + SCALAR_BLOCKS, 256, 0, stream>>>(
      keys_in, values_in, scores_in, ws_tok, out);
}